// TNNWormCore_52999896433145
// MI455X (gfx1250) — compile-verified
//
#include <hip/hip_runtime.h>
#include <math.h>

typedef __attribute__((ext_vector_type(2))) float v2f;
typedef __attribute__((ext_vector_type(8))) float v8f;

#define TWO_PI_F 6.2831853071795864769f
#define NB1 512   // pass-1 block count (fixed -> deterministic reduction)

// ws layout (floats):
//   [0..1]    : final sum|lo_n| for n=0,1
//   [2..9]    : b1' (folded hidden bias, 8)
//   [16..79]  : A-image for W1'   (32 lanes x float2)
//   [80..143] : A-image for W_o2 K-chunk0
//   [144..207]: A-image for W_o2 K-chunk1
//   [208..239]: raw W1' (8x4) for tail kernel
//   [256..]   : pass-1 partials (NB1 x 2)

__device__ __forceinline__ float gelu_exact(float x) {
    return 0.5f * x * (1.0f + erff(x * 0.70710678118654752440f));
}
__device__ __forceinline__ float sigmoidf_(float x) {
    return 1.0f / (1.0f + __expf(-x));
}

// Per-row scalar pipeline: x[6] -> h_r[4] (gelu(LN(torsion(gelu(x W_in^T+b)))))
__device__ __forceinline__ void row_to_hr(const float* __restrict__ xr,
    const float* __restrict__ W_in, const float* __restrict__ b_in,
    const float* __restrict__ r_w,  const float* __restrict__ r_b,
    const float* __restrict__ r_tf,
    float sc0, float sc1, float sig_rtc, float h[4])
{
    float xv[6];
    #pragma unroll
    for (int j = 0; j < 6; ++j) xv[j] = xr[j];
    float a[4];
    #pragma unroll
    for (int k = 0; k < 4; ++k) {
        float s = b_in[k];
        #pragma unroll
        for (int j = 0; j < 6; ++j) s += W_in[k*6 + j] * xv[j];
        a[k] = gelu_exact(s);
    }
    float t[4];
    #pragma unroll
    for (int k = 0; k < 4; ++k) {
        float s = r_b[k];
        #pragma unroll
        for (int j = 0; j < 4; ++j) s += r_w[k*4 + j] * a[j];
        t[k] = s;
    }
    #pragma unroll
    for (int n = 0; n < 2; ++n) {
        const float sc   = (n == 0) ? sc0 : sc1;
        const float invn = (n == 0) ? 1.0f : 0.5f;
        #pragma unroll
        for (int k = 0; k < 4; ++k) {
            float lo = 0.0f;
            #pragma unroll
            for (int j = 0; j < 4; ++j) lo += r_tf[(n*4 + k)*4 + j] * a[j];
            t[k] += sig_rtc * __sinf(lo * sc) * lo * invn;
        }
    }
    float mu = 0.25f * (t[0] + t[1] + t[2] + t[3]);
    float var = 0.0f;
    #pragma unroll
    for (int k = 0; k < 4; ++k) { float d = t[k] - mu; var += d * d; }
    var *= 0.25f;
    float rs = rsqrtf(var + 1e-5f);
    #pragma unroll
    for (int k = 0; k < 4; ++k) h[k] = gelu_exact((t[k] - mu) * rs);
}

// ---------------- prep: constant h_i path, fold cross-flow, build WMMA A-images ---
__global__ void prep_kernel(const float* i_w, const float* i_b, const float* i_tf,
                            const float* i_tc, const float* W_ri, const float* b_ri,
                            const float* W_ir, const float* b_ir, const float* flow_gate,
                            const float* memv, const float* W_o1, const float* b_o1,
                            const float* W_o2, float* ws)
{
    __shared__ float sW1p[32];
    __shared__ float sb1p[8];
    const int lane = threadIdx.x;
    if (lane == 0) {
        const float gate = sigmoidf_(flow_gate[0]);
        const float sitc = sigmoidf_(i_tc[0]);
        float xi[16], base[16], corr[16];
        for (int k = 0; k < 16; ++k) { xi[k] = memv[k]; corr[k] = 0.0f; }
        for (int k = 0; k < 16; ++k) {
            float s = i_b[k];
            for (int j = 0; j < 16; ++j) s += i_w[k*16 + j] * xi[j];
            base[k] = s;
        }
        for (int n = 0; n < 3; ++n) {
            float lo[16]; float ma = 0.0f;
            for (int k = 0; k < 16; ++k) {
                float s = 0.0f;
                for (int j = 0; j < 16; ++j) s += i_tf[(n*16 + k)*16 + j] * xi[j];
                lo[k] = s; ma += fabsf(s);
            }
            const float sc = TWO_PI_F * (float)(n + 1) / (ma * (1.0f/16.0f) + 1e-8f);
            for (int k = 0; k < 16; ++k)
                corr[k] += __sinf(lo[k] * sc) * lo[k] / (float)(n + 1);
        }
        float t[16], mu = 0.0f;
        for (int k = 0; k < 16; ++k) { t[k] = base[k] + sitc * corr[k]; mu += t[k]; }
        mu *= (1.0f/16.0f);
        float var = 0.0f;
        for (int k = 0; k < 16; ++k) { float d = t[k] - mu; var += d * d; }
        var *= (1.0f/16.0f);
        const float rs = rsqrtf(var + 1e-5f);
        float hi0[16];
        for (int k = 0; k < 16; ++k) hi0[k] = gelu_exact((t[k] - mu) * rs);

        // h_r_final = h_r*M + c   (affine fold of the cross-flow)
        float c[4], M[4][4];
        for (int j = 0; j < 4; ++j) {
            float s = 0.0f;
            for (int m = 0; m < 16; ++m) s += (hi0[m] + gate * b_ri[m]) * W_ir[j*16 + m];
            c[j] = gate * s + gate * b_ir[j];
        }
        const float g2 = gate * gate;
        for (int k = 0; k < 4; ++k)
            for (int j = 0; j < 4; ++j) {
                float s = 0.0f;
                for (int m = 0; m < 16; ++m) s += W_ri[m*4 + k] * W_ir[j*16 + m];
                M[k][j] = ((k == j) ? 1.0f : 0.0f) + g2 * s;
            }
        // fold M,c into first output layer: W1'[h][k], b1'[h]
        for (int hh = 0; hh < 8; ++hh) {
            float bb = b_o1[hh];
            for (int j = 0; j < 4; ++j) bb += W_o1[hh*4 + j] * c[j];
            sb1p[hh] = bb;
            for (int k = 0; k < 4; ++k) {
                float s = 0.0f;
                for (int j = 0; j < 4; ++j) s += W_o1[hh*4 + j] * M[k][j];
                sW1p[hh*4 + k] = s;
            }
        }
    }
    __syncthreads();
    // A-matrix lane images: lane holds M = lane&15, K = 2*(lane>>4) + v
    const int m  = lane & 15;
    const int k0 = (lane >> 4) * 2;
    ws[16 + lane*2 + 0] = (m < 8) ? sW1p[m*4 + k0    ] : 0.0f;
    ws[16 + lane*2 + 1] = (m < 8) ? sW1p[m*4 + k0 + 1] : 0.0f;
    ws[80  + lane*2 + 0] = (m < 4) ? W_o2[m*8 + k0    ] : 0.0f;
    ws[80  + lane*2 + 1] = (m < 4) ? W_o2[m*8 + k0 + 1] : 0.0f;
    ws[144 + lane*2 + 0] = (m < 4) ? W_o2[m*8 + k0 + 4] : 0.0f;
    ws[144 + lane*2 + 1] = (m < 4) ? W_o2[m*8 + k0 + 5] : 0.0f;
    if (lane < 8)  ws[2 + lane]   = sb1p[lane];
    if (lane < 32) ws[208 + lane] = sW1p[lane];
}

// ---------------- pass 1: global sum of |lo_n| over the batch --------------------
__global__ void pass1_kernel(const float* __restrict__ x, const float* W_in,
                             const float* b_in, const float* r_tf, int B,
                             float* __restrict__ partials)
{
    __shared__ float s0a[256], s1a[256];
    float s0 = 0.0f, s1 = 0.0f;
    for (int r = blockIdx.x * 256 + threadIdx.x; r < B; r += NB1 * 256) {
        const float* xr = x + (long long)r * 6;
        float xv[6];
        #pragma unroll
        for (int j = 0; j < 6; ++j) xv[j] = xr[j];
        float a[4];
        #pragma unroll
        for (int k = 0; k < 4; ++k) {
            float s = b_in[k];
            #pragma unroll
            for (int j = 0; j < 6; ++j) s += W_in[k*6 + j] * xv[j];
            a[k] = gelu_exact(s);
        }
        #pragma unroll
        for (int n = 0; n < 2; ++n) {
            float acc = 0.0f;
            #pragma unroll
            for (int k = 0; k < 4; ++k) {
                float lo = 0.0f;
                #pragma unroll
                for (int j = 0; j < 4; ++j) lo += r_tf[(n*4 + k)*4 + j] * a[j];
                acc += fabsf(lo);
            }
            if (n == 0) s0 += acc; else s1 += acc;
        }
    }
    s0a[threadIdx.x] = s0; s1a[threadIdx.x] = s1;
    __syncthreads();
    for (int off = 128; off > 0; off >>= 1) {
        if (threadIdx.x < off) {
            s0a[threadIdx.x] += s0a[threadIdx.x + off];
            s1a[threadIdx.x] += s1a[threadIdx.x + off];
        }
        __syncthreads();
    }
    if (threadIdx.x == 0) {
        partials[blockIdx.x*2 + 0] = s0a[0];
        partials[blockIdx.x*2 + 1] = s1a[0];
    }
}

__global__ void reduce_kernel(float* ws)
{
    __shared__ float s0a[256], s1a[256];
    const float* partials = ws + 256;
    float s0 = 0.0f, s1 = 0.0f;
    for (int i = threadIdx.x; i < NB1; i += 256) {
        s0 += partials[2*i]; s1 += partials[2*i + 1];
    }
    s0a[threadIdx.x] = s0; s1a[threadIdx.x] = s1;
    __syncthreads();
    for (int off = 128; off > 0; off >>= 1) {
        if (threadIdx.x < off) {
            s0a[threadIdx.x] += s0a[threadIdx.x + off];
            s1a[threadIdx.x] += s1a[threadIdx.x + off];
        }
        __syncthreads();
    }
    if (threadIdx.x == 0) { ws[0] = s0a[0]; ws[1] = s1a[0]; }
}

// ---------------- pass 2: scalar pipeline + WMMA output projection ---------------
__global__ void __launch_bounds__(256)
pass2_kernel(const float* __restrict__ x, float* __restrict__ out,
             const float* W_in, const float* b_in,
             const float* r_w,  const float* r_b, const float* r_tf, const float* r_tc,
             const float* b_o2, const float* __restrict__ ws, int B)
{
    const int lane = threadIdx.x & 31;
    const int row  = blockIdx.x * 256 + threadIdx.x;   // own row; full tiles only

    // wave-uniform scalars (scalar loads)
    const float denom = 4.0f * (float)B;
    const float sc0 = TWO_PI_F * 1.0f / (ws[0] / denom + 1e-8f);
    const float sc1 = TWO_PI_F * 2.0f / (ws[1] / denom + 1e-8f);
    const float sig_rtc = sigmoidf_(r_tc[0]);

    float h[4];
    row_to_hr(x + (long long)row * 6, W_in, b_in, r_w, r_b, r_tf, sc0, sc1, sig_rtc, h);

    // WMMA operand images (per-lane 8B loads from L2-resident ws)
    const v2f a1  = *(const v2f*)(ws + 16  + lane*2);
    const v2f a20 = *(const v2f*)(ws + 80  + lane*2);
    const v2f a21 = *(const v2f*)(ws + 144 + lane*2);
    float b1p[8];
    #pragma unroll
    for (int i = 0; i < 8; ++i) b1p[i] = ws[2 + i];

    const float lowMask = (lane < 16) ? 1.0f : 0.0f;
    v8f c2init = {};
    c2init[0] = b_o2[0] * lowMask;  c2init[1] = b_o2[1] * lowMask;
    c2init[2] = b_o2[2] * lowMask;  c2init[3] = b_o2[3] * lowMask;

    float og0[4], og1[4];
    #pragma unroll
    for (int g = 0; g < 2; ++g) {
        // B operand of matmul #1: B[K=feat][N=row], rows g*16 + (lane&15)
        const int src = (lane & 15) + (g << 4);
        const float s0 = __shfl(h[0], src), s1v = __shfl(h[1], src);
        const float s2 = __shfl(h[2], src), s3v = __shfl(h[3], src);
        v2f bB; bB.x = (lane < 16) ? s0 : s2; bB.y = (lane < 16) ? s1v : s3v;

        v8f zero = {};
        v8f d1 = __builtin_amdgcn_wmma_f32_16x16x4_f32(
            false, a1, false, bB, (short)0, zero, false, false);

        float act[8];
        #pragma unroll
        for (int i = 0; i < 8; ++i) act[i] = gelu_exact(d1[i] + b1p[i]);

        // B operand of matmul #2 (two K=4 chunks of K=8)
        const float t2 = __shfl(act[2], lane & 15), t3 = __shfl(act[3], lane & 15);
        const float t6 = __shfl(act[6], lane & 15), t7 = __shfl(act[7], lane & 15);
        v2f b20, b21;
        b20.x = (lane < 16) ? act[0] : t2;  b20.y = (lane < 16) ? act[1] : t3;
        b21.x = (lane < 16) ? act[4] : t6;  b21.y = (lane < 16) ? act[5] : t7;

        v8f d2 = __builtin_amdgcn_wmma_f32_16x16x4_f32(
            false, a20, false, b20, (short)0, c2init, false, false);
        d2 = __builtin_amdgcn_wmma_f32_16x16x4_f32(
            false, a21, false, b21, (short)0, d2, false, false);

        if (g == 0) { og0[0]=d2[0]; og0[1]=d2[1]; og0[2]=d2[2]; og0[3]=d2[3]; }
        else        { og1[0]=d2[0]; og1[1]=d2[1]; og1[2]=d2[2]; og1[3]=d2[3]; }
    }

    // merge both groups so every lane stores its own row -> one coalesced b128
    const float u0 = __shfl(og1[0], lane & 15), u1 = __shfl(og1[1], lane & 15);
    const float u2 = __shfl(og1[2], lane & 15), u3 = __shfl(og1[3], lane & 15);
    float4 o;
    o.x = (lane < 16) ? og0[0] : u0;  o.y = (lane < 16) ? og0[1] : u1;
    o.z = (lane < 16) ? og0[2] : u2;  o.w = (lane < 16) ? og0[3] : u3;
    *(float4*)(out + (long long)row * 4) = o;
}

// ---------------- scalar tail (B % 256 rows; keeps WMMA waves at full EXEC) ------
__global__ void tail_kernel(const float* __restrict__ x, float* __restrict__ out,
                            const float* W_in, const float* b_in,
                            const float* r_w, const float* r_b, const float* r_tf,
                            const float* r_tc, const float* W_o2, const float* b_o2,
                            const float* ws, int start, int B)
{
    const int row = start + blockIdx.x * blockDim.x + threadIdx.x;
    if (row >= B) return;
    const float denom = 4.0f * (float)B;
    const float sc0 = TWO_PI_F * 1.0f / (ws[0] / denom + 1e-8f);
    const float sc1 = TWO_PI_F * 2.0f / (ws[1] / denom + 1e-8f);
    const float sig_rtc = sigmoidf_(r_tc[0]);
    float h[4];
    row_to_hr(x + (long long)row * 6, W_in, b_in, r_w, r_b, r_tf, sc0, sc1, sig_rtc, h);
    const float* W1p = ws + 208;
    const float* b1p = ws + 2;
    float o[4];
    #pragma unroll
    for (int q = 0; q < 4; ++q) o[q] = b_o2[q];
    #pragma unroll
    for (int hh = 0; hh < 8; ++hh) {
        float s = b1p[hh];
        #pragma unroll
        for (int k = 0; k < 4; ++k) s += W1p[hh*4 + k] * h[k];
        const float a8 = gelu_exact(s);
        #pragma unroll
        for (int q = 0; q < 4; ++q) o[q] += W_o2[q*8 + hh] * a8;
    }
    #pragma unroll
    for (int q = 0; q < 4; ++q) out[(long long)row*4 + q] = o[q];
}

extern "C" void kernel_launch(void* const* d_in, const int* in_sizes, int n_in,
                              void* d_out, int out_size, void* d_ws, size_t ws_size,
                              hipStream_t stream)
{
    const float* x    = (const float*)d_in[0];
    const float* W_in = (const float*)d_in[1];
    const float* b_in = (const float*)d_in[2];
    const float* r_w  = (const float*)d_in[3];
    const float* r_b  = (const float*)d_in[4];
    const float* r_tf = (const float*)d_in[5];
    const float* r_tc = (const float*)d_in[6];
    const float* i_w  = (const float*)d_in[7];
    const float* i_b  = (const float*)d_in[8];
    const float* i_tf = (const float*)d_in[9];
    const float* i_tc = (const float*)d_in[10];
    const float* W_ri = (const float*)d_in[11];
    const float* b_ri = (const float*)d_in[12];
    const float* W_ir = (const float*)d_in[13];
    const float* b_ir = (const float*)d_in[14];
    const float* fg   = (const float*)d_in[15];
    const float* memv = (const float*)d_in[16];
    const float* W_o1 = (const float*)d_in[17];
    const float* b_o1 = (const float*)d_in[18];
    const float* W_o2 = (const float*)d_in[19];
    const float* b_o2 = (const float*)d_in[20];
    float* out = (float*)d_out;
    float* ws  = (float*)d_ws;

    const int B = in_sizes[0] / 6;
    float* partials = ws + 256;

    prep_kernel<<<1, 32, 0, stream>>>(i_w, i_b, i_tf, i_tc, W_ri, b_ri, W_ir, b_ir,
                                      fg, memv, W_o1, b_o1, W_o2, ws);
    pass1_kernel<<<NB1, 256, 0, stream>>>(x, W_in, b_in, r_tf, B, partials);
    reduce_kernel<<<1, 256, 0, stream>>>(ws);

    const int full = (B / 256) * 256;
    if (full > 0) {
        pass2_kernel<<<full / 256, 256, 0, stream>>>(x, out, W_in, b_in, r_w, r_b,
                                                     r_tf, r_tc, b_o2, ws, B);
    }
    if (full < B) {
        const int rem = B - full;
        tail_kernel<<<(rem + 255) / 256, 256, 0, stream>>>(x, out, W_in, b_in, r_w, r_b,
                                                           r_tf, r_tc, W_o2, b_o2, ws,
                                                           full, B);
    }
}